// GCN_8761733284692
// MI455X (gfx1250) — compile-verified
//
#include <hip/hip_runtime.h>

typedef float v2f __attribute__((ext_vector_type(2)));
typedef float v8f __attribute__((ext_vector_type(8)));

#define N_NODES 10000
#define N_EDGES 320000
#define D_IN    256
#define D_HID   256
#define D_OUT   64

// ---------------------------------------------------------------------------
// Degree: deg[dst] += 1 for every edge, plus one self-loop per node.
// ---------------------------------------------------------------------------
__global__ void gcn_degree(const int* __restrict__ dst, float* __restrict__ deg,
                           int nE, int nN) {
    int i = blockIdx.x * blockDim.x + threadIdx.x;
    int tot = nE + nN;
    if (i >= tot) return;
    int d = (i < nE) ? dst[i] : (i - nE);
    atomicAdd(&deg[d], 1.0f);
}

// deg -> deg^{-1/2} in place (deg >= 1 always due to self-loops)
__global__ void gcn_dinv(float* __restrict__ deg, int n) {
    int i = blockIdx.x * blockDim.x + threadIdx.x;
    if (i >= n) return;
    float d = deg[i];
    deg[i] = (d > 0.0f) ? rsqrtf(d) : 0.0f;
}

// ---------------------------------------------------------------------------
// Dense GEMM C[M,N] = A[M,K] * B[K,N] in full fp32 on the matrix pipe.
// One wave32 computes one 16x16 tile via V_WMMA_F32_16X16X4_F32 (K-step 4).
// A frag: lane<16 -> A[m0+lane][k..k+1], lane>=16 -> A[m0+lane-16][k+2..k+3]
// B frag: lane<16 -> B[k+0/k+1][n0+lane], lane>=16 -> B[k+2/k+3][n0+lane-16]
// C/D   : VGPR r, lane<16 -> C[m0+r][n0+lane], lane>=16 -> C[m0+r+8][...]
// M, N, K are multiples of 16 / 16 / 4 (exact here: 10000, 256|64, 256).
// ---------------------------------------------------------------------------
__global__ void gcn_gemm_wmma_f32(const float* __restrict__ A,
                                  const float* __restrict__ B,
                                  float* __restrict__ C,
                                  int M, int N, int K) {
    const int lane  = threadIdx.x & 31;
    const int tile  = (int)((blockIdx.x * blockDim.x + threadIdx.x) >> 5);
    const int ntile = N >> 4;
    const int mtile = M >> 4;
    if (tile >= mtile * ntile) return;

    const int m0 = (tile / ntile) << 4;
    const int n0 = (tile % ntile) << 4;
    const int mr = lane & 15;            // row/col within tile for this lane
    const int kh = (lane >> 4) << 1;     // 0 for lanes 0-15, 2 for lanes 16-31

    const float* ap = A + (size_t)(m0 + mr) * K + kh;   // contiguous float2 per k-step
    const float* bp = B + (size_t)kh * N + n0 + mr;     // two rows, stride N apart

    v8f acc = {0.f, 0.f, 0.f, 0.f, 0.f, 0.f, 0.f, 0.f};

#pragma unroll 4
    for (int k = 0; k < K; k += 4) {
        v2f a;
        a.x = ap[k];
        a.y = ap[k + 1];
        v2f b;
        b.x = bp[(size_t)k * N];
        b.y = bp[(size_t)k * N + N];
        // (neg_a, A, neg_b, B, c_mod, C, reuse_a, reuse_b)
        acc = __builtin_amdgcn_wmma_f32_16x16x4_f32(false, a, false, b,
                                                    (short)0, acc, false, false);
    }

    const int row0 = m0 + ((lane >> 4) << 3);   // m0 or m0+8
    float* cp = C + (size_t)row0 * N + n0 + mr;
#pragma unroll
    for (int r = 0; r < 8; ++r)
        cp[(size_t)r * N] = acc[r];
}

// ---------------------------------------------------------------------------
// Edge aggregation: out[dst] += dinv[src]*dinv[dst] * h[src]   (scatter-add)
// D/4 threads per edge, each handling 4 consecutive features (float4 gather).
// Self-loops are edge ids >= nE.
// ---------------------------------------------------------------------------
__global__ void gcn_aggregate(const int* __restrict__ src,
                              const int* __restrict__ dst,
                              const float* __restrict__ dinv,
                              const float* __restrict__ h,
                              float* __restrict__ out,
                              int nE, int nN, int D) {
    const int tpe = D >> 2;  // threads per edge
    long long idx = (long long)blockIdx.x * blockDim.x + threadIdx.x;
    long long tot = (long long)(nE + nN) * tpe;
    if (idx >= tot) return;

    int e = (int)(idx / tpe);
    int f = ((int)(idx - (long long)e * tpe)) << 2;

    int s, t;
    if (e < nE) { s = src[e]; t = dst[e]; }
    else        { s = t = e - nE; }

    float w = dinv[s] * dinv[t];
    const float4 v = *(const float4*)(h + (size_t)s * D + f);
    float* o = out + (size_t)t * D + f;
    atomicAdd(o + 0, v.x * w);
    atomicAdd(o + 1, v.y * w);
    atomicAdd(o + 2, v.z * w);
    atomicAdd(o + 3, v.w * w);
}

// h[i] = relu(h[i] + bias[i & mask])   (mask = D-1, D power of two)
__global__ void gcn_bias_relu(float* __restrict__ h, const float* __restrict__ bias,
                              int n, int mask) {
    int i = blockIdx.x * blockDim.x + threadIdx.x;
    if (i >= n) return;
    float v = h[i] + bias[i & mask];
    h[i] = (v > 0.0f) ? v : 0.0f;
}

// out[i] = bias[i & mask]  (seed the final scatter target with b2)
__global__ void gcn_seed_bias(float* __restrict__ out, const float* __restrict__ bias,
                              int n, int mask) {
    int i = blockIdx.x * blockDim.x + threadIdx.x;
    if (i >= n) return;
    out[i] = bias[i & mask];
}

// ---------------------------------------------------------------------------
extern "C" void kernel_launch(void* const* d_in, const int* in_sizes, int n_in,
                              void* d_out, int out_size, void* d_ws, size_t ws_size,
                              hipStream_t stream) {
    (void)in_sizes; (void)n_in; (void)out_size; (void)ws_size;

    const float* x  = (const float*)d_in[0];
    const int*   ei = (const int*)d_in[1];      // [2, E]: row 0 = src, row 1 = dst
    const float* W1 = (const float*)d_in[2];
    const float* b1 = (const float*)d_in[3];
    const float* W2 = (const float*)d_in[4];
    const float* b2 = (const float*)d_in[5];
    float* out = (float*)d_out;

    const int* src = ei;
    const int* dst = ei + N_EDGES;

    // Workspace layout (floats): deg/dinv [N] | h1 [N*256] | agg1 [N*256]
    float* deg  = (float*)d_ws;
    float* h1   = deg + N_NODES;
    float* agg1 = h1 + (size_t)N_NODES * D_HID;

    hipMemsetAsync(deg, 0, N_NODES * sizeof(float), stream);
    hipMemsetAsync(agg1, 0, (size_t)N_NODES * D_HID * sizeof(float), stream);

    // 1) degrees (incl. self-loops), then dinv = deg^{-1/2}
    {
        int tot = N_EDGES + N_NODES;
        gcn_degree<<<(tot + 255) / 256, 256, 0, stream>>>(dst, deg, N_EDGES, N_NODES);
    }
    gcn_dinv<<<(N_NODES + 255) / 256, 256, 0, stream>>>(deg, N_NODES);

    // 2) layer-1 GEMM: h1 = x @ W1   (10000x256 @ 256x256), wmma fp32
    {
        int tiles = (N_NODES / 16) * (D_HID / 16);      // 625*16 = 10000 waves
        int blocks = (tiles + 7) / 8;                   // 8 wave32 per 256-thread block
        gcn_gemm_wmma_f32<<<blocks, 256, 0, stream>>>(x, W1, h1, N_NODES, D_HID, D_IN);
    }

    // 3) layer-1 aggregation: agg1[dst] += norm * h1[src]
    {
        long long tot = (long long)(N_EDGES + N_NODES) * (D_HID / 4);
        gcn_aggregate<<<(unsigned)((tot + 255) / 256), 256, 0, stream>>>(
            src, dst, deg, h1, agg1, N_EDGES, N_NODES, D_HID);
    }

    // 4) z = relu(agg1 + b1), in place
    gcn_bias_relu<<<(N_NODES * D_HID + 255) / 256, 256, 0, stream>>>(
        agg1, b1, N_NODES * D_HID, D_HID - 1);

    // 5) layer-2 GEMM: h2 = z @ W2   (10000x256 @ 256x64) -> reuse h1 buffer
    {
        int tiles = (N_NODES / 16) * (D_OUT / 16);      // 625*4 = 2500 waves
        int blocks = (tiles + 7) / 8;
        gcn_gemm_wmma_f32<<<blocks, 256, 0, stream>>>(agg1, W2, h1, N_NODES, D_OUT, D_HID);
    }

    // 6) seed output with bias b2, then layer-2 aggregation into d_out
    gcn_seed_bias<<<(N_NODES * D_OUT + 255) / 256, 256, 0, stream>>>(
        out, b2, N_NODES * D_OUT, D_OUT - 1);
    {
        long long tot = (long long)(N_EDGES + N_NODES) * (D_OUT / 4);
        gcn_aggregate<<<(unsigned)((tot + 255) / 256), 256, 0, stream>>>(
            src, dst, deg, h1, out, N_EDGES, N_NODES, D_OUT);
    }
}